// Graph_27350351741245
// MI455X (gfx1250) — compile-verified
//
#include <hip/hip_runtime.h>

typedef __attribute__((ext_vector_type(2))) float v2f;
typedef __attribute__((ext_vector_type(8))) float v8f;

#define N_VARS 50000
#define N_TX   200000
#define DD     32

// ---------------------------------------------------------------------------
// Kernel 0: zero the accumulator (d_out doubles as the preds buffer).
// ---------------------------------------------------------------------------
__global__ __launch_bounds__(256) void zero_f32(float* __restrict__ p, int n) {
    int i = blockIdx.x * blockDim.x + threadIdx.x;
    if (i < n) p[i] = 0.0f;
}

// ---------------------------------------------------------------------------
// Kernel 1: one wave per edge (grid-strided).
//   y = W[e] @ x + b[e] via 16x V_WMMA_F32_16X16X4_F32, computed as
//   y^T = x^T * W^T with A rows broadcast (all rows of C identical == y).
//   Scatter with per-lane float atomics (coalesced 128B per edge).
// ---------------------------------------------------------------------------
__global__ __launch_bounds__(256) void edge_wmma(
    const float* __restrict__ theta,
    const float* __restrict__ W,
    const float* __restrict__ bias,
    const int*   __restrict__ src,
    const int*   __restrict__ tgt,
    float*       __restrict__ preds) {

    const int lane = threadIdx.x & 31;
    const int wid  = (blockIdx.x * blockDim.x + threadIdx.x) >> 5;
    const int nw   = (gridDim.x * blockDim.x) >> 5;
    const int l15  = lane & 15;
    const int hi   = (lane >> 4) << 1;   // 0 for lanes 0-15, 2 for lanes 16-31

    for (int e = wid; e < N_TX; e += nw) {
        const float* We = W + (size_t)e * (DD * DD);

        // Prefetch next edge's W tile (4KB): lane*128B covers the whole tile.
        if (e + nw < N_TX) {
            __builtin_prefetch((const void*)(W + (size_t)(e + nw) * (DD * DD) + lane * DD),
                               0, 3);
        }

        const int s = src[e];
        const float xv = theta[s * DD + lane];          // coalesced 128B gather

        // C init: bias broadcast into every row (all rows of C stay identical).
        const float bv0 = bias[e * DD + l15];           // outputs 0..15
        const float bv1 = bias[e * DD + 16 + l15];      // outputs 16..31
        v8f c0, c1;
        #pragma unroll
        for (int i = 0; i < 8; ++i) { c0[i] = bv0; c1[i] = bv1; }

        #pragma unroll
        for (int k = 0; k < 8; ++k) {
            // A(16x4): every row m holds x[4k .. 4k+3].
            // Lane layout: VGPR0 = K=(4k+hi), VGPR1 = K=(4k+hi+1).
            v2f a;
            a.x = __shfl(xv, 4 * k + hi,     32);
            a.y = __shfl(xv, 4 * k + hi + 1, 32);

            // B(4x16) = W^T tile: B[kk][n] = W[nblk*16+n][4k+kk].
            // Lane l supplies (n=l&15, kk=hi..hi+1): contiguous float2 of row n.
            v2f b0 = *(const v2f*)(We + (size_t)(l15)      * DD + 4 * k + hi);
            v2f b1 = *(const v2f*)(We + (size_t)(16 + l15) * DD + 4 * k + hi);

            c0 = __builtin_amdgcn_wmma_f32_16x16x4_f32(
                     false, a, false, b0, (short)0, c0, false, false);
            c1 = __builtin_amdgcn_wmma_f32_16x16x4_f32(
                     false, a, false, b1, (short)0, c1, false, false);
        }

        // All rows of C are identical, so component 0 of every lane holds y[n]:
        //   lanes 0-15:  c0[0] = y[lane];   lanes 16-31: c1[0] = y[lane].
        const float y = (lane < 16) ? c0[0] : c1[0];

        const int t = tgt[e];
        unsafeAtomicAdd(&preds[t * DD + lane], y);      // global_atomic_add_f32
    }
}

// ---------------------------------------------------------------------------
// Kernel 2: resid = theta - preds (in place over d_out), float4 vectorized.
// ---------------------------------------------------------------------------
__global__ __launch_bounds__(256) void resid_f32(
    const float4* __restrict__ theta4, float4* __restrict__ out4, int n4) {
    int i = blockIdx.x * blockDim.x + threadIdx.x;
    if (i < n4) {
        float4 t = theta4[i];
        float4 p = out4[i];
        float4 r;
        r.x = t.x - p.x; r.y = t.y - p.y; r.z = t.z - p.z; r.w = t.w - p.w;
        out4[i] = r;
    }
}

extern "C" void kernel_launch(void* const* d_in, const int* in_sizes, int n_in,
                              void* d_out, int out_size, void* d_ws, size_t ws_size,
                              hipStream_t stream) {
    const float* theta = (const float*)d_in[0];
    const float* W     = (const float*)d_in[1];
    const float* b     = (const float*)d_in[2];
    const int*   src   = (const int*)d_in[3];
    const int*   tgt   = (const int*)d_in[4];
    float* out = (float*)d_out;

    const int n = N_VARS * DD;     // == out_size == 1,600,000

    zero_f32<<<(n + 255) / 256, 256, 0, stream>>>(out, n);
    edge_wmma<<<2048, 256, 0, stream>>>(theta, W, b, src, tgt, out);
    const int n4 = n / 4;
    resid_f32<<<(n4 + 255) / 256, 256, 0, stream>>>(
        (const float4*)theta, (float4*)out, n4);
}